// SparseAttention_74852690035424
// MI455X (gfx1250) — compile-verified
//
#include <hip/hip_runtime.h>
#include <math.h>

// Problem constants (match reference)
#define NB   2
#define LQ   1024
#define LK   2048
#define DIM  512
#define H    8
#define HD   64
#define TOPK 204

typedef __attribute__((ext_vector_type(2))) float v2f;
typedef __attribute__((ext_vector_type(8))) float v8f;

// ---------------------------------------------------------------------------
// CDNA5 native fp32 WMMA: D(16x16) = A(16x4) * B(4x16) + C
// A frag (per lane): VGPR0=K{0 | lanes0-15, 2 | lanes16-31}, VGPR1=K{1,3}
// -> lane (half,l) loads X[row=l][k+2*half .. k+2*half+1] as float2 (same for B)
// C/D: VGPR r holds (M=r | lanes0-15, M=8+r | lanes16-31), N = lane&15
// ---------------------------------------------------------------------------
__device__ __forceinline__ v8f wmma4(v2f a, v2f b, v8f c) {
  return __builtin_amdgcn_wmma_f32_16x16x4_f32(
      /*neg_a=*/false, a, /*neg_b=*/false, b,
      /*c_mod=*/(short)0, c, /*reuse_a=*/false, /*reuse_b=*/false);
}

// Monotone float -> uint key (total order, -inf smallest)
__device__ __forceinline__ unsigned fkey(float f) {
  unsigned u = __float_as_uint(f);
  return (u & 0x80000000u) ? ~u : (u | 0x80000000u);
}

// ---------------------------------------------------------------------------
// C[m][j] = sum_k X[m][k] * W[j][k] + bias[j]   (torch Linear: X @ W.T + b)
// grid = (M/16, DIM/16), block = 32 (one wave per 16x16 output tile)
// ---------------------------------------------------------------------------
__global__ void linear_wmma(const float* __restrict__ X,
                            const float* __restrict__ W,
                            const float* __restrict__ bias,
                            float* __restrict__ C) {
  const int l    = threadIdx.x & 15;
  const int half = threadIdx.x >> 4;
  const int m0   = blockIdx.x * 16;
  const int n0   = blockIdx.y * 16;

  const float bval = bias[n0 + l];
  v8f acc = {bval, bval, bval, bval, bval, bval, bval, bval};

  const float* xr = X + (size_t)(m0 + l) * DIM + 2 * half;
  const float* wr = W + (size_t)(n0 + l) * DIM + 2 * half;
#pragma unroll 4
  for (int k = 0; k < DIM; k += 4) {
    v2f a = *(const v2f*)(xr + k);
    v2f b = *(const v2f*)(wr + k);
    acc = wmma4(a, b, acc);
  }

  float* cr = C + (size_t)m0 * DIM + n0 + l;
#pragma unroll
  for (int r = 0; r < 8; ++r)
    cr[(size_t)(r + 8 * half) * DIM] = acc[r];
}

// ---------------------------------------------------------------------------
// Fused: per-head scores (WMMA) -> LDS -> exact radix top-K -> softmax ->
// weighted V gather. One block = one (n, h, 16-query tile). 8 waves.
// ---------------------------------------------------------------------------
#define SC_STRIDE  (LK + 1)                         // bank stagger
#define SMEM_SC_B  (16 * SC_STRIDE * 4)             // 131136 B
#define SMEM_HIST_B (8 * 256 * 4)                   // 8192 B
#define SMEM_LIST_B (16 * TOPK * 4)                 // 13056 B
#define SMEM_CNT_B  (16 * 2 * 4)                    // 128 B
#define SMEM_BYTES  (SMEM_SC_B + SMEM_HIST_B + SMEM_LIST_B + SMEM_CNT_B)

__global__ void sparse_attn_wmma(const float* __restrict__ Qp,
                                 const float* __restrict__ Kp,
                                 const float* __restrict__ inF,
                                 const unsigned char* __restrict__ mask,
                                 float* __restrict__ attn) {
  extern __shared__ char smem_raw[];
  float* sc   = (float*)smem_raw;                               // [16][SC_STRIDE]
  int*   hist = (int*)(smem_raw + SMEM_SC_B);                   // [8][256]
  int*   list = (int*)(smem_raw + SMEM_SC_B + SMEM_HIST_B);     // [16][TOPK]
  int*   cnt  = (int*)(smem_raw + SMEM_SC_B + SMEM_HIST_B + SMEM_LIST_B); // [16][2]

  const int tid  = threadIdx.x;
  const int w    = tid >> 5;      // wave 0..7
  const int lane = tid & 31;
  const int l    = lane & 15;
  const int half = lane >> 4;

  const int QT = LQ / 16;
  const int b  = blockIdx.x;
  const int qt = b % QT;
  const int h  = (b / QT) % H;
  const int n  = b / (QT * H);
  const int q0 = qt * 16;
  const int hbase = h * HD;

  // ---------------- phase 1: score tile (16 queries x LK keys) --------------
  // Cache this wave's Q fragments for the full K=64 head dim (16 steps of 4).
  v2f qa[16];
  {
    const float* qrow = Qp + ((size_t)n * LQ + q0 + l) * DIM + hbase + 2 * half;
#pragma unroll
    for (int s = 0; s < 16; ++s) qa[s] = *(const v2f*)(qrow + 4 * s);
  }
  for (int t = 0; t < 16; ++t) {            // 16 key tiles of 16 per wave
    const int key0 = w * 256 + t * 16;
    const float* krow = Kp + ((size_t)n * LK + key0 + l) * DIM + hbase + 2 * half;
    v8f c = {0.f, 0.f, 0.f, 0.f, 0.f, 0.f, 0.f, 0.f};
#pragma unroll
    for (int s = 0; s < 16; ++s) {
      v2f bb = *(const v2f*)(krow + 4 * s);
      c = wmma4(qa[s], bb, c);
    }
    const bool msk = mask[(size_t)n * LK + key0 + l] != 0;
#pragma unroll
    for (int r = 0; r < 8; ++r)
      sc[(r + 8 * half) * SC_STRIDE + key0 + l] = msk ? -INFINITY : c[r];
  }
  __syncthreads();

  // ---------------- phase 2: per-row top-K + softmax + V gather -------------
  for (int rr = 0; rr < 2; ++rr) {          // each wave owns rows w and w+8
    const int row = w + 8 * rr;
    const float* srow = sc + row * SC_STRIDE;

    // Row max (== max over selected set, since top-1 is always selected)
    float m = -INFINITY;
    for (int i = 0; i < LK / 32; ++i) m = fmaxf(m, srow[lane + 32 * i]);
#pragma unroll
    for (int off = 16; off >= 1; off >>= 1) m = fmaxf(m, __shfl_xor(m, off, 32));

    // Exact 4-pass MSB-first radix select: threshold = TOPK-th largest key
    unsigned prefix = 0;
    int kneed = TOPK;
    int* hw = hist + w * 256;               // per-wave histogram (no cross-wave races)
    for (int pass = 3; pass >= 0; --pass) {
      __syncthreads();
      for (int bb = lane; bb < 256; bb += 32) hw[bb] = 0;
      __syncthreads();
      const int sh = 8 * pass;
      for (int i = 0; i < LK / 32; ++i) {
        unsigned u = fkey(srow[lane + 32 * i]);
        bool ok = (pass == 3) || ((u >> (sh + 8)) == prefix);
        if (ok) atomicAdd(&hw[(u >> sh) & 255u], 1);
      }
      __syncthreads();
      int cum = 0, chosen = 0;
      for (int bb = 255; bb >= 0; --bb) {   // all lanes scan redundantly (LDS bcast)
        int cb = hw[bb];
        if (cum + cb >= kneed) { chosen = bb; break; }
        cum += cb;
      }
      kneed -= cum;
      prefix = (prefix << 8) | (unsigned)chosen;
    }
    const unsigned Tu = prefix;             // key of the TOPK-th largest score
    // kneed = number of Tu-valued ties still to include

    // Build selected index list + softmax partition function Z
    int* cr = cnt + row * 2;
    if (lane == 0) { atomicExch(&cr[0], 0); atomicExch(&cr[1], 0); }
    int* lrow = list + row * TOPK;
    float zp = 0.0f;
    for (int i = 0; i < LK / 32; ++i) {
      const int key = lane + 32 * i;
      const float v = srow[key];
      const unsigned u = fkey(v);
      bool sel = (u > Tu);
      if (!sel && u == Tu) {
        int t = atomicAdd(&cr[1], 1);       // claim a tie slot
        sel = (t < kneed);
      }
      if (sel) {
        int p = atomicAdd(&cr[0], 1);
        lrow[p] = key;
        zp += __expf(v - m);
      }
    }
#pragma unroll
    for (int off = 16; off >= 1; off >>= 1) zp += __shfl_xor(zp, off, 32);
    const float invZ = 1.0f / zp;           // softmax + redundant renorm == /Z

    // Cooperative weighted gather: lane owns output dims {2*lane, 2*lane+1}
    const float* vbase = inF + (size_t)n * LK * DIM + hbase + 2 * lane;
    float a0 = 0.f, a1 = 0.f;
    for (int i = 0; i < TOPK; ++i) {
      const int key = lrow[i];              // LDS broadcast read
      const float wgt = __expf(srow[key] - m) * invZ;
      v2f vv = *(const v2f*)(vbase + (size_t)key * DIM);
      a0 += wgt * vv[0];
      a1 += wgt * vv[1];
    }
    float* orow = attn + ((size_t)n * LQ + q0 + row) * DIM + hbase + 2 * lane;
    orow[0] = a0;
    orow[1] = a1;
  }
}

// ---------------------------------------------------------------------------
extern "C" void kernel_launch(void* const* d_in, const int* in_sizes, int n_in,
                              void* d_out, int out_size, void* d_ws, size_t ws_size,
                              hipStream_t stream) {
  (void)in_sizes; (void)n_in; (void)out_size; (void)ws_size;
  const float*         query = (const float*)d_in[0];
  const float*         inF   = (const float*)d_in[1];
  const unsigned char* maskb = (const unsigned char*)d_in[2];
  const float*         Wq    = (const float*)d_in[3];
  const float*         bq    = (const float*)d_in[4];
  const float*         Wk    = (const float*)d_in[5];
  const float*         bk    = (const float*)d_in[6];
  const float*         Wo    = (const float*)d_in[7];
  const float*         bo    = (const float*)d_in[8];
  float*               out   = (float*)d_out;

  // Workspace: Qp (4MB) | Kp (8MB) | attn (4MB)
  float* Qp   = (float*)d_ws;
  float* Kp   = Qp + (size_t)NB * LQ * DIM;
  float* attn = Kp + (size_t)NB * LK * DIM;

  // Opt into >64KB dynamic LDS (CDNA5 WGP has 320KB). Deterministic, not
  // stream-ordered, safe under graph capture.
  (void)hipFuncSetAttribute((const void*)sparse_attn_wmma,
                            hipFuncAttributeMaxDynamicSharedMemorySize,
                            SMEM_BYTES);

  const dim3 wblk(32);
  // Q and K projections
  linear_wmma<<<dim3(NB * LQ / 16, DIM / 16), wblk, 0, stream>>>(query, Wq, bq, Qp);
  linear_wmma<<<dim3(NB * LK / 16, DIM / 16), wblk, 0, stream>>>(inF,   Wk, bk, Kp);
  // Fused scores + top-k + softmax + V gather
  sparse_attn_wmma<<<dim3(NB * H * (LQ / 16)), dim3(256), SMEM_BYTES, stream>>>(
      Qp, Kp, inF, maskb, attn);
  // Output projection
  linear_wmma<<<dim3(NB * LQ / 16, DIM / 16), wblk, 0, stream>>>(attn, Wo, bo, out);
}